// JKNet_14491219656877
// MI455X (gfx1250) — compile-verified
//
#include <hip/hip_runtime.h>

typedef __attribute__((ext_vector_type(2))) float v2f;
typedef __attribute__((ext_vector_type(8))) float v8f;

#define HDIM 16

// ---------------------------------------------------------------------------
// Degree / normalization precompute
// ---------------------------------------------------------------------------
__global__ void k_deg_init(float* __restrict__ deg, int n) {
    int i = blockIdx.x * blockDim.x + threadIdx.x;
    if (i < n) deg[i] = 1.0f;  // self-loop contribution
}

__global__ void k_deg_edges(const long long* __restrict__ dst,
                            float* __restrict__ deg, int e) {
    int i = blockIdx.x * blockDim.x + threadIdx.x;
    if (i < e) atomicAdd(&deg[(int)dst[i]], 1.0f);
}

__global__ void k_dinv(const float* __restrict__ deg, float* __restrict__ dinv, int n) {
    int i = blockIdx.x * blockDim.x + threadIdx.x;
    if (i < n) {
        float d = deg[i];
        dinv[i] = (d > 0.0f) ? rsqrtf(d) : 0.0f;
    }
}

__global__ void k_norm(const long long* __restrict__ src,
                       const long long* __restrict__ dst,
                       const float* __restrict__ dinv,
                       float* __restrict__ norm, int e) {
    int i = blockIdx.x * blockDim.x + threadIdx.x;
    if (i < e) norm[i] = dinv[(int)src[i]] * dinv[(int)dst[i]];
}

// ---------------------------------------------------------------------------
// Dense transform via V_WMMA_F32_16X16X4_F32 (fp32 matrix path; GEMM is
// ~0.5 GFLOP total so full f32 precision costs nothing here).
// One wave handles a 16-node tile: t = in_tile @ W, fully unrolled over K
// (compile-time). Tail step (K=58 only) uses clamped unconditional loads +
// value selects so EXEC never diverges (WMMA requires EXEC all-ones and the
// previous guarded-load formulation made clang emit saveexec branches).
//
// Wave32 layouts (ISA 7.12.2):
//   A (16x4 f32): lanes 0-15 -> M=lane, K={k0,k0+1}; lanes 16-31 -> K={k0+2,k0+3}
//   B (4x16 f32): lanes 0-15 -> N=lane, K rows {k0,k0+1}; lanes 16-31 -> {k0+2,k0+3}
//   C/D (16x16 f32): 8 VGPRs; lanes 0-15 -> M=r, lanes 16-31 -> M=r+8; N=lane&15
// Also seeds agg = t*dinv^2 + bias (self-loop message + bias fused).
// ---------------------------------------------------------------------------
template <int K, int STRIDE>
__global__ __launch_bounds__(256)
void k_transform_wmma(const float* __restrict__ in,
                      const float* __restrict__ W,     // K x 16 row-major
                      const float* __restrict__ bias,
                      const float* __restrict__ dinv,
                      float* __restrict__ t, float* __restrict__ agg,
                      int ntiles) {
    int wave = (blockIdx.x * blockDim.x + threadIdx.x) >> 5;  // wave-uniform
    if (wave >= ntiles) return;                               // whole-wave exit
    int lane  = threadIdx.x & 31;
    int half  = lane >> 4;      // 0 | 1
    int l15   = lane & 15;
    int kb    = half * 2;       // lane's K-pair base within the 4-wide step
    int node0 = wave * 16;

    const float* arow = in + (size_t)(node0 + l15) * STRIDE;

    v8f acc = {};
    constexpr int FULL = K / 4;            // steps with all K indices in-bounds
    constexpr bool TAIL = (K % 4) != 0;    // K=58 -> one partial step (K%4==2)

#pragma unroll
    for (int s = 0; s < FULL; ++s) {
        int k0 = s * 4 + kb;
        float2 av = *(const float2*)(arow + k0);          // global_load_b64
        v2f a, b;
        a.x = av.x;
        a.y = av.y;
        b.x = W[(size_t)(k0    ) * HDIM + l15];
        b.y = W[(size_t)(k0 + 1) * HDIM + l15];
        acc = __builtin_amdgcn_wmma_f32_16x16x4_f32(
                  /*neg_a=*/false, a, /*neg_b=*/false, b,
                  /*c_mod=*/(short)0, acc, /*reuse_a=*/false, /*reuse_b=*/false);
    }
    if constexpr (TAIL) {
        // K%4==2: each lane's pair is entirely in-bounds (half 0) or entirely
        // out (half 1). Clamp address, load unconditionally, select to zero.
        int  k0 = FULL * 4 + kb;
        bool v  = (k0 < K);
        int  kc = v ? k0 : 0;
        float2 av = *(const float2*)(arow + kc);
        float  bx = W[(size_t)(kc    ) * HDIM + l15];
        float  by = W[(size_t)(kc + 1) * HDIM + l15];
        v2f a, b;
        a.x = v ? av.x : 0.0f;
        a.y = v ? av.y : 0.0f;
        b.x = v ? bx   : 0.0f;
        b.y = v ? by   : 0.0f;
        acc = __builtin_amdgcn_wmma_f32_16x16x4_f32(
                  false, a, false, b, (short)0, acc, false, false);
    }

    int   mbase = half * 8;
    float bcol  = bias[l15];
#pragma unroll
    for (int r = 0; r < 8; ++r) {
        int   node = node0 + mbase + r;
        float v    = acc[r];
        t[(size_t)node * HDIM + l15] = v;
        float di = dinv[node];
        agg[(size_t)node * HDIM + l15] = v * di * di + bcol;  // self-loop + bias
    }
}

// ---------------------------------------------------------------------------
// Edge propagate: agg[dst] += t[src] * norm  (f32 global atomics; t/agg are
// 6.4MB each so gather + atomics stay resident in MI455X's 192MB L2)
// ---------------------------------------------------------------------------
__global__ __launch_bounds__(256)
void k_scatter(const long long* __restrict__ src, const long long* __restrict__ dst,
               const float* __restrict__ norm, const float* __restrict__ t,
               float* __restrict__ agg, int e) {
    int i = blockIdx.x * blockDim.x + threadIdx.x;
    if (i >= e) return;
    int   s   = (int)src[i];
    int   d   = (int)dst[i];
    float nrm = norm[i];
    const float4* ts = (const float4*)(t + (size_t)s * HDIM);
    float4 v0 = ts[0], v1 = ts[1], v2 = ts[2], v3 = ts[3];
    float* ag = agg + (size_t)d * HDIM;
    atomicAdd(ag + 0,  v0.x * nrm); atomicAdd(ag + 1,  v0.y * nrm);
    atomicAdd(ag + 2,  v0.z * nrm); atomicAdd(ag + 3,  v0.w * nrm);
    atomicAdd(ag + 4,  v1.x * nrm); atomicAdd(ag + 5,  v1.y * nrm);
    atomicAdd(ag + 6,  v1.z * nrm); atomicAdd(ag + 7,  v1.w * nrm);
    atomicAdd(ag + 8,  v2.x * nrm); atomicAdd(ag + 9,  v2.y * nrm);
    atomicAdd(ag + 10, v2.z * nrm); atomicAdd(ag + 11, v2.w * nrm);
    atomicAdd(ag + 12, v3.x * nrm); atomicAdd(ag + 13, v3.y * nrm);
    atomicAdd(ag + 14, v3.z * nrm); atomicAdd(ag + 15, v3.w * nrm);
}

// ---------------------------------------------------------------------------
// ReLU + JumpingKnowledge running max
// ---------------------------------------------------------------------------
__global__ void k_finalize(const float* __restrict__ agg, float* __restrict__ h,
                           float* __restrict__ jk, int total, int first) {
    int i = blockIdx.x * blockDim.x + threadIdx.x;
    if (i >= total) return;
    float v = fmaxf(agg[i], 0.0f);
    h[i] = v;
    jk[i] = first ? v : fmaxf(jk[i], v);
}

// ---------------------------------------------------------------------------
// out = jk @ fc_w + fc_b   ([n,16] x [16,1])
// ---------------------------------------------------------------------------
__global__ void k_output(const float* __restrict__ jk, const float* __restrict__ fc_w,
                         const float* __restrict__ fc_b, float* __restrict__ out, int n) {
    int i = blockIdx.x * blockDim.x + threadIdx.x;
    if (i >= n) return;
    const float4* row = (const float4*)(jk + (size_t)i * HDIM);
    const float4* w   = (const float4*)fc_w;
    float4 r0 = row[0], r1 = row[1], r2 = row[2], r3 = row[3];
    float4 w0 = w[0],   w1 = w[1],   w2 = w[2],   w3 = w[3];
    float acc = r0.x * w0.x + r0.y * w0.y + r0.z * w0.z + r0.w * w0.w
              + r1.x * w1.x + r1.y * w1.y + r1.z * w1.z + r1.w * w1.w
              + r2.x * w2.x + r2.y * w2.y + r2.z * w2.z + r2.w * w2.w
              + r3.x * w3.x + r3.y * w3.y + r3.z * w3.z + r3.w * w3.w;
    out[i] = acc + fc_b[0];
}

// ---------------------------------------------------------------------------
extern "C" void kernel_launch(void* const* d_in, const int* in_sizes, int n_in,
                              void* d_out, int out_size, void* d_ws, size_t ws_size,
                              hipStream_t stream) {
    const float*     x    = (const float*)d_in[0];
    const long long* ei   = (const long long*)d_in[1];  // int64 edge_index [2,E]
    const float*     W0   = (const float*)d_in[2];      // 58 x 16
    const float*     Ws   = (const float*)d_in[3];      // 9 x 16 x 16
    const float*     bs   = (const float*)d_in[4];      // 10 x 16
    const float*     fc_w = (const float*)d_in[5];      // 16 x 1
    const float*     fc_b = (const float*)d_in[6];      // 1
    float*           out  = (float*)d_out;

    const int F_IN = 58, L = 10;
    const int n = in_sizes[0] / F_IN;
    const int e = in_sizes[1] / 2;
    const long long* src = ei;
    const long long* dst = ei + e;

    // workspace layout (floats)
    float* ws   = (float*)d_ws;
    size_t n16  = (size_t)n * HDIM;
    float* t    = ws;               // n*16
    float* agg  = t    + n16;       // n*16
    float* h    = agg  + n16;       // n*16
    float* jk   = h    + n16;       // n*16
    float* deg  = jk   + n16;       // n
    float* dinv = deg  + n;         // n
    float* norm = dinv + n;         // e

    const int TB = 256;
    int gN   = (n + TB - 1) / TB;
    int gE   = (e + TB - 1) / TB;
    int gNH  = ((int)n16 + TB - 1) / TB;
    int ntiles = (n + 15) / 16;
    int gT   = (ntiles + (TB / 32) - 1) / (TB / 32);  // 8 tiles (waves) per block

    // --- normalization precompute (once; reused by all 10 layers) ---
    k_deg_init <<<gN, TB, 0, stream>>>(deg, n);
    k_deg_edges<<<gE, TB, 0, stream>>>(dst, deg, e);
    k_dinv     <<<gN, TB, 0, stream>>>(deg, dinv, n);
    k_norm     <<<gE, TB, 0, stream>>>(src, dst, dinv, norm, e);

    // --- layer 0: x(58) -> h(16) ---
    k_transform_wmma<58, 58><<<gT, TB, 0, stream>>>(x, W0, bs, dinv, t, agg, ntiles);
    k_scatter            <<<gE, TB, 0, stream>>>(src, dst, norm, t, agg, e);
    k_finalize           <<<gNH, TB, 0, stream>>>(agg, h, jk, (int)n16, /*first=*/1);

    // --- layers 1..9: h(16) -> h(16) ---
    for (int l = 1; l < L; ++l) {
        const float* W = Ws + (size_t)(l - 1) * HDIM * HDIM;
        const float* b = bs + (size_t)l * HDIM;
        k_transform_wmma<16, 16><<<gT, TB, 0, stream>>>(h, W, b, dinv, t, agg, ntiles);
        k_scatter            <<<gE, TB, 0, stream>>>(src, dst, norm, t, agg, e);
        k_finalize           <<<gNH, TB, 0, stream>>>(agg, h, jk, (int)n16, /*first=*/0);
    }

    // --- readout ---
    k_output<<<gN, TB, 0, stream>>>(jk, fc_w, fc_b, out, n);
}